// TopKGate_38336878084276
// MI455X (gfx1250) — compile-verified
//
#include <hip/hip_runtime.h>

typedef float v2f __attribute__((ext_vector_type(2)));
typedef float v8f __attribute__((ext_vector_type(8)));

#define S_TOK 16384
#define DIMK  4096
#define NE    64
#define TOPK  2

// ---------------------------------------------------------------------------
// Kernel 0: zero the 128-float workspace (importance[64] + load[64]).
// ---------------------------------------------------------------------------
__global__ void zero_ws_kernel(float* ws) {
    ws[threadIdx.x] = 0.0f;
}

// ---------------------------------------------------------------------------
// Kernel 1: fused GEMM (WMMA f32 16x16x4) + softmax + top-2 + stats.
// One wave = 16 tokens x 64 experts. Block = 256 threads = 8 waves.
// ---------------------------------------------------------------------------
__launch_bounds__(256)
__global__ void gate_kernel(const float* __restrict__ x,
                            const float* __restrict__ W,
                            const float* __restrict__ bias,
                            float* __restrict__ outIdx,   // S*K floats
                            float* __restrict__ outVal,   // S*K floats
                            float* __restrict__ gImp,     // 64 floats
                            float* __restrict__ gLoad)    // 64 floats
{
    __shared__ float ldsImp[NE];
    __shared__ float ldsLoad[NE];

    const int tid = threadIdx.x;
    if (tid < NE) { ldsImp[tid] = 0.0f; ldsLoad[tid] = 0.0f; }
    __syncthreads();

    const int lane = tid & 31;
    const int wave = tid >> 5;
    const int gw   = blockIdx.x * 8 + wave;   // global wave index, 0..1023
    const int t0   = gw * 16;                 // first token of this wave
    const int n    = lane & 15;               // sub-lane (row/col within tile)
    const int hi   = lane >> 4;               // half-wave select
    const int koff = hi * 2;                  // K offset per documented f32 A/B layout

    // A: x rows t0..t0+15 (lane n -> row n; both halves read same rows, K split)
    const float* xrow = x + (size_t)(t0 + n) * DIMK + koff;
    // B: W rows for 4 expert groups {16G + n}
    const float* wr0 = W + (size_t)(n +  0) * DIMK + koff;
    const float* wr1 = W + (size_t)(n + 16) * DIMK + koff;
    const float* wr2 = W + (size_t)(n + 32) * DIMK + koff;
    const float* wr3 = W + (size_t)(n + 48) * DIMK + koff;

    v8f acc[4] = {};   // 4 tiles of 16x16 f32: experts [0,16), [16,32), [32,48), [48,64)

    for (int k0 = 0; k0 < DIMK; k0 += 32) {
        // stream-ahead prefetch of the x row (one cacheline ahead)
        __builtin_prefetch(xrow + k0 + 32, 0, 3);
#pragma unroll
        for (int k = k0; k < k0 + 32; k += 4) {
            float2 af = *(const float2*)(xrow + k);
            float2 b0 = *(const float2*)(wr0 + k);
            float2 b1 = *(const float2*)(wr1 + k);
            float2 b2 = *(const float2*)(wr2 + k);
            float2 b3 = *(const float2*)(wr3 + k);
            v2f a  = {af.x, af.y};
            v2f v0 = {b0.x, b0.y};
            v2f v1 = {b1.x, b1.y};
            v2f v2 = {b2.x, b2.y};
            v2f v3 = {b3.x, b3.y};
            acc[0] = __builtin_amdgcn_wmma_f32_16x16x4_f32(false, a, false, v0,
                                                           (short)0, acc[0], false, false);
            acc[1] = __builtin_amdgcn_wmma_f32_16x16x4_f32(false, a, false, v1,
                                                           (short)0, acc[1], false, false);
            acc[2] = __builtin_amdgcn_wmma_f32_16x16x4_f32(false, a, false, v2,
                                                           (short)0, acc[2], false, false);
            acc[3] = __builtin_amdgcn_wmma_f32_16x16x4_f32(false, a, false, v3,
                                                           (short)0, acc[3], false, false);
        }
    }

    // Bias per expert group for this lane's column
    float bv[4];
#pragma unroll
    for (int G = 0; G < 4; ++G) bv[G] = bias[G * 16 + n];

    float impAcc[4] = {0.f, 0.f, 0.f, 0.f};

    // C/D layout: VGPR g holds token (t0+g) in lanes 0-15 and token (t0+8+g)
    // in lanes 16-31; expert = 16*G + n. Two tokens processed per g in parallel.
#pragma unroll
    for (int g = 0; g < 8; ++g) {
        float p0 = acc[0][g] + bv[0];
        float p1 = acc[1][g] + bv[1];
        float p2 = acc[2][g] + bv[2];
        float p3 = acc[3][g] + bv[3];

        // row max across 16 lanes (xor masks stay within each half-wave)
        float m = fmaxf(fmaxf(p0, p1), fmaxf(p2, p3));
#pragma unroll
        for (int mask = 1; mask < 16; mask <<= 1)
            m = fmaxf(m, __shfl_xor(m, mask, 32));

        float e0 = __expf(p0 - m);
        float e1 = __expf(p1 - m);
        float e2 = __expf(p2 - m);
        float e3 = __expf(p3 - m);
        float s = e0 + e1 + e2 + e3;
#pragma unroll
        for (int mask = 1; mask < 16; mask <<= 1)
            s += __shfl_xor(s, mask, 32);
        float inv = 1.0f / s;

        float q0 = e0 * inv, q1 = e1 * inv, q2 = e2 * inv, q3 = e3 * inv;
        impAcc[0] += q0; impAcc[1] += q1; impAcc[2] += q2; impAcc[3] += q3;

        // lane-local top-2 over the lane's 4 experts
        float v1 = q0; int i1 = n;
        float v2 = -1.0f; int i2 = -1;
        {
            float qv[3] = {q1, q2, q3};
            int   qi[3] = {16 + n, 32 + n, 48 + n};
#pragma unroll
            for (int j = 0; j < 3; ++j) {
                if (qv[j] > v1) { v2 = v1; i2 = i1; v1 = qv[j]; i1 = qi[j]; }
                else if (qv[j] > v2) { v2 = qv[j]; i2 = qi[j]; }
            }
        }
        // cross-lane top-2 merge within the 16-lane half
#pragma unroll
        for (int mask = 1; mask < 16; mask <<= 1) {
            float ov1 = __shfl_xor(v1, mask, 32);
            int   oi1 = __shfl_xor(i1, mask, 32);
            float ov2 = __shfl_xor(v2, mask, 32);
            int   oi2 = __shfl_xor(i2, mask, 32);
            if (ov1 > v1) {
                float nv2; int ni2;
                if (v1 > ov2) { nv2 = v1; ni2 = i1; }
                else          { nv2 = ov2; ni2 = oi2; }
                v1 = ov1; i1 = oi1; v2 = nv2; i2 = ni2;
            } else if (ov1 > v2) {
                v2 = ov1; i2 = oi1;
            }
        }

        if (n == 0) {
            int tok = t0 + g + hi * 8;
            outIdx[tok * 2 + 0] = (float)i1;
            outIdx[tok * 2 + 1] = (float)i2;
            outVal[tok * 2 + 0] = v1;
            outVal[tok * 2 + 1] = v2;
            atomicAdd(&ldsLoad[i1], 1.0f);   // argmax histogram
        }
    }

    // importance partial sums -> LDS -> one global atomic per expert per WG
#pragma unroll
    for (int G = 0; G < 4; ++G)
        atomicAdd(&ldsImp[G * 16 + n], impAcc[G]);
    __syncthreads();
    if (tid < NE) {
        atomicAdd(&gImp[tid],  ldsImp[tid]);
        atomicAdd(&gLoad[tid], ldsLoad[tid]);
    }
}

// ---------------------------------------------------------------------------
// Kernel 2: aux_loss = E * sum( (imp[e]/S) * (load[e]/S) )
// ---------------------------------------------------------------------------
__global__ void finalize_kernel(const float* __restrict__ gImp,
                                const float* __restrict__ gLoad,
                                float* __restrict__ outAux)
{
    __shared__ float red[NE];
    int e = threadIdx.x;
    const float invS = 1.0f / (float)S_TOK;
    red[e] = (gImp[e] * invS) * (gLoad[e] * invS);
    __syncthreads();
    if (e == 0) {
        float s = 0.0f;
        for (int i = 0; i < NE; ++i) s += red[i];
        outAux[0] = (float)NE * s;
    }
}

// ---------------------------------------------------------------------------
extern "C" void kernel_launch(void* const* d_in, const int* in_sizes, int n_in,
                              void* d_out, int out_size, void* d_ws, size_t ws_size,
                              hipStream_t stream) {
    const float* x = (const float*)d_in[0];   // [S, DIM]
    const float* W = (const float*)d_in[1];   // [E, DIM]
    const float* b = (const float*)d_in[2];   // [E]
    float* out = (float*)d_out;               // [S*K idx | S*K vals | aux]
    float* ws  = (float*)d_ws;                // [64 imp | 64 load]

    zero_ws_kernel<<<1, 128, 0, stream>>>(ws);
    gate_kernel<<<S_TOK / 128, 256, 0, stream>>>(
        x, W, b,
        out,                          // indices (as float)
        out + (size_t)S_TOK * TOPK,   // values
        ws, ws + NE);
    finalize_kernel<<<1, NE, 0, stream>>>(ws, ws + NE,
        out + (size_t)2 * S_TOK * TOPK);
}